// FeatureModel_40303973106250
// MI455X (gfx1250) — compile-verified
//
#include <hip/hip_runtime.h>
#include <math.h>

// ---------------------------------------------------------------------------
// FeatureModel pipeline for MI455X (gfx1250, wave32):
//   1) edge_kernel    : r^k powers scatter-added into features[atom][32]
//   2) gram_kernel    : G = F^T F via V_WMMA_F32_16X16X4_F32 + column sums
//   3) eigen_kernel   : top-3 eigenvectors of centered Gram (subspace iter)
//   4) project_kernel : out = F @ V   [n_atoms, 3]
// ---------------------------------------------------------------------------

#define MAX_K 32
#define INV_CUTOFF 0.2f   // 1/5.0

typedef __attribute__((ext_vector_type(2))) float v2f;
typedef __attribute__((ext_vector_type(8))) float v8f;

// ---------------- Phase 0: zero scratch (features + G + colsum) ------------
__global__ void zero_kernel(float* __restrict__ p, long n) {
    long i = (long)blockIdx.x * blockDim.x + threadIdx.x;
    if (i < n) p[i] = 0.0f;
}

// ---------------- Phase 1: per-edge powers, scatter-add --------------------
__global__ void edge_kernel(const float* __restrict__ vec,
                            const int*   __restrict__ first_atom,
                            float*       __restrict__ feat,
                            int nE) {
    int e = blockIdx.x * blockDim.x + threadIdx.x;
    if (e >= nE) return;
    float x = vec[3 * e + 0];
    float y = vec[3 * e + 1];
    float z = vec[3 * e + 2];
    float r = sqrtf(x * x + y * y + z * z) * INV_CUTOFF;
    float* dst = feat + (long)first_atom[e] * MAX_K;
    float p = 1.0f;
#pragma unroll
    for (int k = 0; k < MAX_K; ++k) {
        unsafeAtomicAdd(dst + k, p);   // global_atomic_add_f32, lives in L2
        p *= r;
    }
}

// ---------------- Phase 2: G = F^T F via WMMA f32 16x16x4 ------------------
// G[j,k] = sum_i F[i,j] F[i,k].  Per 4-atom chunk:
//   A-tile (16x4 f32): lanes 0-15 -> M=j, v0=K0,v1=K1; lanes 16-31 -> v0=K2,v1=K3
//   B-tile (4x16 f32): lanes 0-15 -> row K0 (v0), row K1 (v1); lanes 16-31 -> K2,K3
// For a Gram product these per-lane layouts coincide, so one v2f load (P for
// columns 0-15, Q for columns 16-31) feeds both operands of all four tiles.
__global__ void gram_kernel(const float* __restrict__ feat,
                            int nAtoms,
                            float* __restrict__ G,        // 32x32
                            float* __restrict__ colsum) { // 32
    const int lane = threadIdx.x & 31;
    const int half = lane >> 4;     // 0: atoms {a0,a1}, 1: atoms {a2,a3}
    const int l16  = lane & 15;
    const long wave_id = ((long)blockIdx.x * blockDim.x + threadIdx.x) >> 5;
    const long nwaves  = ((long)gridDim.x * blockDim.x) >> 5;

    v8f acc00 = {}, acc01 = {}, acc10 = {}, acc11 = {};
    float cs_lo = 0.0f, cs_hi = 0.0f;

    for (long base = wave_id * 4; base < nAtoms; base += nwaves * 4) {
        long a0 = base + 2 * half;
        long a1 = a0 + 1;
        v2f P = {}, Q = {};
        if (a0 < nAtoms) {
            const float* r0 = feat + a0 * MAX_K;
            P.x = r0[l16];  Q.x = r0[16 + l16];
        }
        if (a1 < nAtoms) {
            const float* r1 = feat + a1 * MAX_K;
            P.y = r1[l16];  Q.y = r1[16 + l16];
        }
        cs_lo += P.x + P.y;     // column l16, this half's two atoms
        cs_hi += Q.x + Q.y;     // column 16+l16
        acc00 = __builtin_amdgcn_wmma_f32_16x16x4_f32(false, P, false, P, (short)0, acc00, false, false);
        acc01 = __builtin_amdgcn_wmma_f32_16x16x4_f32(false, P, false, Q, (short)0, acc01, false, false);
        acc10 = __builtin_amdgcn_wmma_f32_16x16x4_f32(false, Q, false, P, (short)0, acc10, false, false);
        acc11 = __builtin_amdgcn_wmma_f32_16x16x4_f32(false, Q, false, Q, (short)0, acc11, false, false);
    }

    // C/D layout: VGPR v, lanes 0-15 -> row v, lanes 16-31 -> row 8+v; col = l16
#pragma unroll
    for (int v = 0; v < 8; ++v) {
        int row = v + 8 * half;
        unsafeAtomicAdd(&G[(row)      * 32 + l16],      acc00[v]);
        unsafeAtomicAdd(&G[(row)      * 32 + 16 + l16], acc01[v]);
        unsafeAtomicAdd(&G[(16 + row) * 32 + l16],      acc10[v]);
        unsafeAtomicAdd(&G[(16 + row) * 32 + 16 + l16], acc11[v]);
    }
    unsafeAtomicAdd(&colsum[l16],      cs_lo);
    unsafeAtomicAdd(&colsum[16 + l16], cs_hi);
}

// ---------------- Phase 3: top-3 eigenvectors of centered Gram -------------
__device__ inline float wave_sum32(float x) {
#pragma unroll
    for (int o = 16; o > 0; o >>= 1) x += __shfl_xor(x, o, 32);
    return x;
}

__global__ void eigen_kernel(const float* __restrict__ G,
                             const float* __restrict__ colsum,
                             float* __restrict__ Vout,   // 32x3
                             int nAtoms) {
    __shared__ float Gc[32][32];
    __shared__ float V[32][3];
    const int j = threadIdx.x;   // one wave32: lane j owns row j

    // Centered Gram: Gc = G - mu*(cs x 1 + 1 x cs) + N*mu^2
    float cs_j   = colsum[j];
    float total  = wave_sum32(cs_j);
    float mu     = total / ((float)nAtoms * 32.0f);
    float nmu2   = (float)nAtoms * mu * mu;
#pragma unroll
    for (int k = 0; k < 32; ++k) {
        float cs_k = __shfl(cs_j, k, 32);
        Gc[j][k] = G[j * 32 + k] - mu * cs_k - mu * cs_j + nmu2;
    }
    // Deterministic non-degenerate init
#pragma unroll
    for (int c = 0; c < 3; ++c)
        V[j][c] = __sinf(0.37f * (float)((j + 1) * (c + 1)) + 0.11f * (float)c);
    __syncthreads();

    for (int it = 0; it < 200; ++it) {
        // W = Gc @ V  (each lane: its row of W)
        float w0 = 0.f, w1 = 0.f, w2 = 0.f;
#pragma unroll
        for (int k = 0; k < 32; ++k) {
            float g = Gc[j][k];
            w0 += g * V[k][0]; w1 += g * V[k][1]; w2 += g * V[k][2];
        }
        // Modified Gram-Schmidt across lanes (columns live across the wave)
        float n0 = wave_sum32(w0 * w0);
        w0 *= rsqrtf(n0 + 1e-30f);
        float r01 = wave_sum32(w0 * w1);
        float r02 = wave_sum32(w0 * w2);
        w1 -= r01 * w0;  w2 -= r02 * w0;
        float n1 = wave_sum32(w1 * w1);
        w1 *= rsqrtf(n1 + 1e-30f);
        float r12 = wave_sum32(w1 * w2);
        w2 -= r12 * w1;
        float n2 = wave_sum32(w2 * w2);
        w2 *= rsqrtf(n2 + 1e-30f);
        __syncthreads();
        V[j][0] = w0; V[j][1] = w1; V[j][2] = w2;
        __syncthreads();
    }
    Vout[j * 3 + 0] = V[j][0];
    Vout[j * 3 + 1] = V[j][1];
    Vout[j * 3 + 2] = V[j][2];
}

// ---------------- Phase 4: out = F @ V  ------------------------------------
__global__ void project_kernel(const float* __restrict__ feat,
                               const float* __restrict__ V,   // 32x3
                               float* __restrict__ out,
                               int nAtoms) {
    __shared__ float Vs[32][3];
    if (threadIdx.x < 32) {
        Vs[threadIdx.x][0] = V[threadIdx.x * 3 + 0];
        Vs[threadIdx.x][1] = V[threadIdx.x * 3 + 1];
        Vs[threadIdx.x][2] = V[threadIdx.x * 3 + 2];
    }
    __syncthreads();
    int a = blockIdx.x * blockDim.x + threadIdx.x;
    if (a >= nAtoms) return;
    const float* f = feat + (long)a * MAX_K;
    float o0 = 0.f, o1 = 0.f, o2 = 0.f;
#pragma unroll
    for (int k = 0; k < MAX_K; ++k) {
        float x = f[k];
        o0 += x * Vs[k][0]; o1 += x * Vs[k][1]; o2 += x * Vs[k][2];
    }
    out[3 * a + 0] = o0; out[3 * a + 1] = o1; out[3 * a + 2] = o2;
}

// ---------------------------------------------------------------------------
extern "C" void kernel_launch(void* const* d_in, const int* in_sizes, int n_in,
                              void* d_out, int out_size, void* d_ws, size_t ws_size,
                              hipStream_t stream) {
    const float* vectors    = (const float*)d_in[0];
    const int*   first_atom = (const int*)d_in[1];
    // n_atoms (d_in[2]) is a device scalar; derive host-side from out_size.
    const int nEdges = in_sizes[0] / 3;
    const int nAtoms = out_size / 3;

    // Workspace layout (floats)
    float* ws       = (float*)d_ws;
    float* features = ws;                               // nAtoms*32
    float* G        = features + (long)nAtoms * MAX_K;  // 32*32
    float* colsum   = G + 32 * 32;                      // 32
    float* V        = colsum + 32;                      // 32*3
    float* out      = (float*)d_out;

    const long nZero = (long)nAtoms * MAX_K + 32 * 32 + 32;
    {
        int blk = 256;
        long grid = (nZero + blk - 1) / blk;
        zero_kernel<<<(unsigned)grid, blk, 0, stream>>>(ws, nZero);
    }
    {
        int blk = 256;
        int grid = (nEdges + blk - 1) / blk;
        edge_kernel<<<grid, blk, 0, stream>>>(vectors, first_atom, features, nEdges);
    }
    {
        int blk = 256;          // 8 waves per block
        int grid = 256;         // 2048 waves striding the atom chunks
        gram_kernel<<<grid, blk, 0, stream>>>(features, nAtoms, G, colsum);
    }
    eigen_kernel<<<1, 32, 0, stream>>>(G, colsum, V, nAtoms);
    {
        int blk = 256;
        int grid = (nAtoms + blk - 1) / blk;
        project_kernel<<<grid, blk, 0, stream>>>(features, V, out, nAtoms);
    }
}